// RochtaschelAttention_79499844649153
// MI455X (gfx1250) — compile-verified
//
#include <hip/hip_runtime.h>
#include <hip/hip_bf16.h>

// Problem constants (from reference: B=32, S=4096, D=512)
constexpr int B      = 32;
constexpr int S      = 4096;
constexpr int D      = 512;
constexpr int HALF   = S / 2;          // 2048
constexpr int CHUNKS = 64;             // chunks per batch -> 2048 waves total
constexpr int RPW    = HALF / CHUNKS;  // 32 rows per wave
constexpr int WPB    = 4;              // waves per block (128 threads)
constexpr int NBUF   = 4;              // async pipeline depth (rows in flight/wave)
constexpr int PSTRIDE = 516;           // partial stride in floats: m,l,pad,pad,racc[512]

// diag workspace layout (floats)
constexpr int OFF_WY = 0;
constexpr int OFF_WP = 512;
constexpr int OFF_WX = 1024;
constexpr int OFF_WH = 1536;
constexpr int DIAG_FLOATS = 2048;

__device__ __forceinline__ float fast_tanh(float x) {
#if __has_builtin(__builtin_amdgcn_tanhf)
  return __builtin_amdgcn_tanhf(x);
#else
  float e = __expf(2.0f * x);
  return 1.0f - 2.0f / (e + 1.0f);
#endif
}

// Async-copy one 512-float row (2048B): 4 x b128, 32 lanes x 16B each.
// IOFFSET is added to BOTH the LDS address and the global address (ISA 08 §4.4),
// so one LDS-base VGPR + one global-base VGPR pair covers the whole row.
__device__ __forceinline__ void async_row(unsigned lds_base, const float* gbase) {
  asm volatile(
      "global_load_async_to_lds_b128 %0, %1, off\n\t"
      "global_load_async_to_lds_b128 %0, %1, off offset:512\n\t"
      "global_load_async_to_lds_b128 %0, %1, off offset:1024\n\t"
      "global_load_async_to_lds_b128 %0, %1, off offset:1536"
      :: "v"(lds_base), "v"(gbase)
      : "memory");
}

// ---------------- Kernel 0: gather diagonals ----------------
__global__ void k_diag(const float* __restrict__ Wy, const float* __restrict__ Wh,
                       const float* __restrict__ Wp, const float* __restrict__ Wx,
                       float* __restrict__ diag) {
  int d = threadIdx.x;  // 512 threads, 1 block
  diag[OFF_WY + d] = Wy[(size_t)d * (D + 1)];
  diag[OFF_WP + d] = Wp[(size_t)d * (D + 1)];
  diag[OFF_WX + d] = Wx[(size_t)d * (D + 1)];
  diag[OFF_WH + d] = Wh[(size_t)d * (D + 1)];
}

// ---------------- Kernel 1: single-pass online softmax partials ----------------
// One wave handles 32 rows of one (batch, chunk). Rows streamed through a
// per-wave 4-deep LDS ring buffer via async global->LDS copies (ASYNCcnt).
__global__ __launch_bounds__(WPB * 32) void k_partial(const float* __restrict__ inp,
                                                      const float* __restrict__ wvec,
                                                      const float* __restrict__ diag,
                                                      float* __restrict__ part) {
  __shared__ float lds[WPB][NBUF][D];  // 32 KB
  const int wave  = threadIdx.x >> 5;
  const int lane  = threadIdx.x & 31;
  const int gw    = blockIdx.x * WPB + wave;   // 0..2047
  const int b     = gw >> 6;                   // /CHUNKS
  const int chunk = gw & (CHUNKS - 1);

  const float* yrow = inp + ((size_t)b * S + (size_t)chunk * RPW) * D + lane * 4;

  // Per-lane constants: lane owns d = {(k*32+lane)*4 + j}, k=0..3, j=0..3
  float4 wy4[4], hd4[4], wv4[4], racc[4];
  {
    const float4* wy = (const float4*)(diag + OFF_WY);
    const float4* wh = (const float4*)(diag + OFF_WH);
    const float4* wv = (const float4*)wvec;
    const float4* hn = (const float4*)(inp + ((size_t)b * S + (S - 1)) * D);
#pragma unroll
    for (int k = 0; k < 4; ++k) {
      int i = k * 32 + lane;
      wy4[k] = wy[i];
      wv4[k] = wv[i];
      float4 h = hn[i], c = wh[i];
      hd4[k] = make_float4(h.x * c.x, h.y * c.y, h.z * c.z, h.w * c.w);
      racc[k] = make_float4(0.f, 0.f, 0.f, 0.f);
    }
  }

  float m = -3.0e38f, l = 0.0f;

  unsigned lb[NBUF];
#pragma unroll
  for (int i = 0; i < NBUF; ++i)
    lb[i] = (unsigned)(unsigned long long)&lds[wave][i][lane * 4];

  // Prologue: prefetch rows 0..3 (ASYNCcnt = 16)
#pragma unroll
  for (int i = 0; i < NBUF; ++i) async_row(lb[i], yrow + (size_t)i * D);

  for (int s = 0; s < RPW; ++s) {
    // Async loads complete in order: <=12 outstanding => row s has landed.
    if (s + NBUF < RPW) {
      asm volatile("s_wait_asynccnt 12" ::: "memory");
    } else {
      asm volatile("s_wait_asynccnt 0" ::: "memory");
    }

    const float4* src = (const float4*)&lds[wave][s & (NBUF - 1)][0];
    float4 y4[4];
#pragma unroll
    for (int k = 0; k < 4; ++k) y4[k] = src[k * 32 + lane];

    // Ensure ds_load_b128 reads have retired before overwriting this buffer.
    asm volatile("s_wait_dscnt 0" ::: "memory");
    if (s + NBUF < RPW)
      async_row(lb[s & (NBUF - 1)], yrow + (size_t)(s + NBUF) * D);

    // score_s = sum_d w[d] * tanh(y*wy + hn*wh)
    float sc = 0.f;
#pragma unroll
    for (int k = 0; k < 4; ++k) {
      sc += wv4[k].x * fast_tanh(fmaf(y4[k].x, wy4[k].x, hd4[k].x));
      sc += wv4[k].y * fast_tanh(fmaf(y4[k].y, wy4[k].y, hd4[k].y));
      sc += wv4[k].z * fast_tanh(fmaf(y4[k].z, wy4[k].z, hd4[k].z));
      sc += wv4[k].w * fast_tanh(fmaf(y4[k].w, wy4[k].w, hd4[k].w));
    }
    // wave32 butterfly reduction
#pragma unroll
    for (int off = 16; off > 0; off >>= 1) sc += __shfl_xor(sc, off, 32);

    // online softmax update
    float mn   = fmaxf(m, sc);
    float cold = __expf(m - mn);
    float p    = __expf(sc - mn);
    l = l * cold + p;
#pragma unroll
    for (int k = 0; k < 4; ++k) {
      racc[k].x = fmaf(racc[k].x, cold, y4[k].x * p);
      racc[k].y = fmaf(racc[k].y, cold, y4[k].y * p);
      racc[k].z = fmaf(racc[k].z, cold, y4[k].z * p);
      racc[k].w = fmaf(racc[k].w, cold, y4[k].w * p);
    }
    m = mn;
  }

  float* pp = part + (size_t)gw * PSTRIDE;
  if (lane == 0) { pp[0] = m; pp[1] = l; }
  float4* pr = (float4*)(pp + 4);
#pragma unroll
  for (int k = 0; k < 4; ++k) pr[k * 32 + lane] = racc[k];
}

// ---------------- Kernel 2: combine partials + epilogue ----------------
__global__ __launch_bounds__(256) void k_combine(const float* __restrict__ inp,
                                                 const float* __restrict__ diag,
                                                 const float* __restrict__ part,
                                                 float* __restrict__ out) {
  __shared__ float msh[CHUNKS], lsh[CHUNKS], esh[CHUNKS];
  __shared__ float Linv;
  const int b = blockIdx.x, t = threadIdx.x;

  if (t < CHUNKS) {
    const float* pp = part + (size_t)(b * CHUNKS + t) * PSTRIDE;
    msh[t] = pp[0];
    lsh[t] = pp[1];
  }
  __syncthreads();
  if (t == 0) {
    float M = -3.0e38f;
    for (int i = 0; i < CHUNKS; ++i) M = fmaxf(M, msh[i]);
    float L = 0.f;
    for (int i = 0; i < CHUNKS; ++i) {
      float e = __expf(msh[i] - M);
      esh[i] = e;
      L += lsh[i] * e;
    }
    Linv = 1.0f / L;
  }
  __syncthreads();

  const float iL = Linv;
  float rx = 0.f, ry = 0.f;
#pragma unroll 4
  for (int i = 0; i < CHUNKS; ++i) {
    const float2* pr = (const float2*)(part + (size_t)(b * CHUNKS + i) * PSTRIDE + 4);
    float2 v = pr[t];
    float e = esh[i];
    rx = fmaf(e, v.x, rx);
    ry = fmaf(e, v.y, ry);
  }
  const int d = t * 2;
  float2 hn = *(const float2*)(inp + ((size_t)b * S + (S - 1)) * D + d);
  float2 wp = *(const float2*)(diag + OFF_WP + d);
  float2 wx = *(const float2*)(diag + OFF_WX + d);
  out[(size_t)b * D + d]     = fast_tanh(fmaf(rx * iL, wp.x, hn.x * wx.x));
  out[(size_t)b * D + d + 1] = fast_tanh(fmaf(ry * iL, wp.y, hn.y * wx.y));
}

extern "C" void kernel_launch(void* const* d_in, const int* in_sizes, int n_in,
                              void* d_out, int out_size, void* d_ws, size_t ws_size,
                              hipStream_t stream) {
  const float* inputs = (const float*)d_in[0];
  const float* W_y    = (const float*)d_in[1];
  const float* W_h    = (const float*)d_in[2];
  const float* W_p    = (const float*)d_in[3];
  const float* W_x    = (const float*)d_in[4];
  const float* w      = (const float*)d_in[5];

  float* ws   = (float*)d_ws;
  float* diag = ws;                   // 2048 floats
  float* part = ws + DIAG_FLOATS;     // 32*64*516 floats (~4.2 MB)

  k_diag<<<1, D, 0, stream>>>(W_y, W_h, W_p, W_x, diag);
  k_partial<<<(B * CHUNKS) / WPB, WPB * 32, 0, stream>>>(inputs, w, diag, part);
  k_combine<<<B, D / 2, 0, stream>>>(inputs, diag, part, (float*)d_out);
}